// GNN_63256278335755
// MI455X (gfx1250) — compile-verified
//
#include <hip/hip_runtime.h>
#include <hip/hip_bf16.h>

#define HD 128
#define NNODES 50000
#define NEDGES 600000
#define NLAYERS 4
#define KC 32  // K-chunk staged in LDS (32 k-values = 8 WMMA k-steps)

typedef float v2f __attribute__((ext_vector_type(2)));
typedef float v8f __attribute__((ext_vector_type(8)));

static __device__ __forceinline__ v8f vzero8() {
    v8f z;
#pragma unroll
    for (int i = 0; i < 8; ++i) z[i] = 0.f;
    return z;
}

static __device__ __forceinline__ v8f wmma_f32(v2f a, v2f b, v8f c) {
    // D = A(16x4 f32) * B(4x16 f32) + C(16x16 f32)
    return __builtin_amdgcn_wmma_f32_16x16x4_f32(false, a, false, b, (short)0, c,
                                                 false, false);
}

static __device__ __forceinline__ float silu(float x) {
    return x / (1.f + __expf(-x));
}

// Stage a KC x 128 chunk of a row-major weight matrix into LDS in
// WMMA-B-fragment order: sW[p][n] = (W[kc+2p][n], W[kc+2p+1][n]) as float2.
// Must be called by all 128 threads; caller handles barriers.
static __device__ __forceinline__ void stage_weights(v2f* __restrict__ sW,
                                                     const float* __restrict__ W,
                                                     int kc, int tid) {
    const float* wsrc = W + (size_t)kc * HD + tid;  // column tid
#pragma unroll
    for (int p = 0; p < KC / 2; ++p) {
        v2f w;
        w.x = wsrc[(2 * p) * HD];
        w.y = wsrc[(2 * p + 1) * HD];
        sW[p * HD + tid] = w;
    }
}

// ---------------------------------------------------------------- embedding
__global__ __launch_bounds__(256) void embed_kernel(
    const float* __restrict__ loc, const float* __restrict__ vel,
    const float* __restrict__ emb_w, const float* __restrict__ emb_b,
    float* __restrict__ h) {
    int idx = blockIdx.x * blockDim.x + threadIdx.x;  // over N*H
    if (idx >= NNODES * HD) return;
    int n = idx >> 7;
    int c = idx & (HD - 1);
    float x = emb_b[c];
#pragma unroll
    for (int k = 0; k < 3; ++k) x = fmaf(loc[n * 3 + k], emb_w[k * HD + c], x);
#pragma unroll
    for (int k = 0; k < 3; ++k) x = fmaf(vel[n * 3 + k], emb_w[(k + 3) * HD + c], x);
    h[idx] = x;
}

// ---------------------------------------------------------------- zero agg
__global__ __launch_bounds__(256) void zero_kernel(float* __restrict__ p, int count) {
    int i = blockIdx.x * blockDim.x + threadIdx.x;
    if (i < count) p[i] = 0.f;
}

// ---------------------------------------------------------------- edge model
// 4 waves/block, one 16-edge tile per wave; weights staged through LDS.
// m = silu(silu([h[row],h[col],ea] @ W1 + b1) @ W2 + b2); atomicAdd into agg[row].
__global__ __launch_bounds__(128) void edge_kernel(
    const float* __restrict__ h, const float* __restrict__ edge_attr,
    const int* __restrict__ row, const int* __restrict__ col,
    const float* __restrict__ W1,  // 258 x 128 (this layer)
    const float* __restrict__ b1,  // 128
    const float* __restrict__ W2,  // 128 x 128
    const float* __restrict__ b2,  // 128
    float* __restrict__ agg) {
    __shared__ v2f sW[(KC / 2) * HD];   // 16KB: weight chunk, fragment layout
    __shared__ float sT[4][16 * HD];    // 32KB: per-wave GEMM1 output staging
    const int tid  = threadIdx.x;
    const int lane = tid & 31;
    const int wave = tid >> 5;
    const int tile = blockIdx.x * 4 + wave;  // exact: NEDGES/16 % 4 == 0
    const int base = tile * 16;
    const int mrow = lane & 15;        // A-fragment row this lane owns
    const int kgrp = (lane >> 4) * 2;  // K sub-offset (0 or 2)
    const int khalf = lane >> 4;       // which fragment pair
    const int mo   = 8 * (lane >> 4);  // C-fragment row offset
    const int ncol = lane & 15;        // C/B column within 16-tile

    const int r_idx = row[base + mrow];
    const int c_idx = col[base + mrow];
    const float* hr = h + (size_t)r_idx * HD;
    const float* hc = h + (size_t)c_idx * HD;

    // ---- GEMM1: [16 x 256] @ [256 x 128]
    v8f acc[8];
#pragma unroll
    for (int nt = 0; nt < 8; ++nt) acc[nt] = vzero8();

    for (int kc = 0; kc < 2 * HD; kc += KC) {
        __syncthreads();  // previous chunk fully consumed
        stage_weights(sW, W1, kc, tid);
        __syncthreads();  // chunk visible to all waves
#pragma unroll
        for (int s = 0; s < KC / 4; ++s) {
            const int kb = kc + 4 * s + kgrp;
            const float* src = (kb < HD) ? (hr + kb) : (hc + (kb - HD));
            v2f a = *(const v2f*)src;  // 8B-aligned (kb even)
            const v2f* bp = &sW[(2 * s + khalf) * HD + ncol];
#pragma unroll
            for (int nt = 0; nt < 8; ++nt) {
                acc[nt] = wmma_f32(a, bp[nt * 16], acc[nt]);
            }
        }
    }

    // rank-2 edge_attr term + bias + SiLU, stage to LDS in [m][n] layout
    float ea0[8], ea1[8];
#pragma unroll
    for (int g = 0; g < 8; ++g) {
        const int e = base + g + mo;
        ea0[g] = edge_attr[2 * e];
        ea1[g] = edge_attr[2 * e + 1];
    }
    const float* w1e0 = W1 + (size_t)256 * HD;
    const float* w1e1 = W1 + (size_t)257 * HD;
    float* sTw = sT[wave];
#pragma unroll
    for (int nt = 0; nt < 8; ++nt) {
        const int n = nt * 16 + ncol;
        const float we0 = w1e0[n], we1 = w1e1[n], bb = b1[n];
#pragma unroll
        for (int g = 0; g < 8; ++g) {
            float x = acc[nt][g] + ea0[g] * we0 + ea1[g] * we1 + bb;
            sTw[(g + mo) * HD + n] = silu(x);
        }
    }
    asm volatile("s_wait_dscnt 0" ::: "memory");  // wave-local LDS RAW

    // ---- GEMM2: [16 x 128] @ [128 x 128]
    v8f acc2[8];
#pragma unroll
    for (int nt = 0; nt < 8; ++nt) acc2[nt] = vzero8();

    for (int kc = 0; kc < HD; kc += KC) {
        __syncthreads();
        stage_weights(sW, W2, kc, tid);
        __syncthreads();
#pragma unroll
        for (int s = 0; s < KC / 4; ++s) {
            const int kb = kc + 4 * s + kgrp;
            v2f a = *(const v2f*)&sTw[mrow * HD + kb];
            const v2f* bp = &sW[(2 * s + khalf) * HD + ncol];
#pragma unroll
            for (int nt = 0; nt < 8; ++nt) {
                acc2[nt] = wmma_f32(a, bp[nt * 16], acc2[nt]);
            }
        }
    }

    // bias + SiLU + atomic scatter-add into agg[row]
    int rg[8];
#pragma unroll
    for (int g = 0; g < 8; ++g) rg[g] = row[base + g + mo];
#pragma unroll
    for (int nt = 0; nt < 8; ++nt) {
        const int n = nt * 16 + ncol;
        const float bb = b2[n];
#pragma unroll
        for (int g = 0; g < 8; ++g) {
            float x = silu(acc2[nt][g] + bb);
            atomicAdd(&agg[(size_t)rg[g] * HD + n], x);
        }
    }
}

// ---------------------------------------------------------------- node model
// h[n] += silu([h[n],agg[n]] @ W1 + b1) @ W2 + b2
__global__ __launch_bounds__(128) void node_kernel(
    float* __restrict__ h, const float* __restrict__ agg,
    const float* __restrict__ W1,  // 256 x 128
    const float* __restrict__ b1, const float* __restrict__ W2,
    const float* __restrict__ b2) {
    __shared__ v2f sW[(KC / 2) * HD];
    __shared__ float sT[4][16 * HD];
    const int tid  = threadIdx.x;
    const int lane = tid & 31;
    const int wave = tid >> 5;
    const int node_tiles = NNODES / 16;  // 3125
    const int tile = blockIdx.x * 4 + wave;
    const bool active = tile < node_tiles;        // wave-uniform
    const int tt = active ? tile : node_tiles - 1;  // clamp: keep barriers uniform
    const int base = tt * 16;
    const int mrow = lane & 15;
    const int kgrp = (lane >> 4) * 2;
    const int khalf = lane >> 4;
    const int mo   = 8 * (lane >> 4);
    const int ncol = lane & 15;

    const float* hr = h + (size_t)(base + mrow) * HD;
    const float* ar = agg + (size_t)(base + mrow) * HD;

    v8f acc[8];
#pragma unroll
    for (int nt = 0; nt < 8; ++nt) acc[nt] = vzero8();

    for (int kc = 0; kc < 2 * HD; kc += KC) {
        __syncthreads();
        stage_weights(sW, W1, kc, tid);
        __syncthreads();
#pragma unroll
        for (int s = 0; s < KC / 4; ++s) {
            const int kb = kc + 4 * s + kgrp;
            const float* src = (kb < HD) ? (hr + kb) : (ar + (kb - HD));
            v2f a = *(const v2f*)src;
            const v2f* bp = &sW[(2 * s + khalf) * HD + ncol];
#pragma unroll
            for (int nt = 0; nt < 8; ++nt) {
                acc[nt] = wmma_f32(a, bp[nt * 16], acc[nt]);
            }
        }
    }

    float* sTw = sT[wave];
#pragma unroll
    for (int nt = 0; nt < 8; ++nt) {
        const int n = nt * 16 + ncol;
        const float bb = b1[n];
#pragma unroll
        for (int g = 0; g < 8; ++g) {
            sTw[(g + mo) * HD + n] = silu(acc[nt][g] + bb);
        }
    }
    asm volatile("s_wait_dscnt 0" ::: "memory");

    v8f acc2[8];
#pragma unroll
    for (int nt = 0; nt < 8; ++nt) acc2[nt] = vzero8();

    for (int kc = 0; kc < HD; kc += KC) {
        __syncthreads();
        stage_weights(sW, W2, kc, tid);
        __syncthreads();
#pragma unroll
        for (int s = 0; s < KC / 4; ++s) {
            const int kb = kc + 4 * s + kgrp;
            v2f a = *(const v2f*)&sTw[mrow * HD + kb];
            const v2f* bp = &sW[(2 * s + khalf) * HD + ncol];
#pragma unroll
            for (int nt = 0; nt < 8; ++nt) {
                acc2[nt] = wmma_f32(a, bp[nt * 16], acc2[nt]);
            }
        }
    }

    // residual update (no SiLU on second linear); rows are tile-exclusive
    if (active) {
#pragma unroll
        for (int nt = 0; nt < 8; ++nt) {
            const int n = nt * 16 + ncol;
            const float bb = b2[n];
#pragma unroll
            for (int g = 0; g < 8; ++g) {
                float* dst = &h[(size_t)(base + g + mo) * HD + n];
                *dst = *dst + (acc2[nt][g] + bb);
            }
        }
    }
}

// ---------------------------------------------------------------- decoder
__global__ __launch_bounds__(128) void decode_kernel(
    const float* __restrict__ h, const float* __restrict__ dw1,
    const float* __restrict__ db1, const float* __restrict__ dw2,
    const float* __restrict__ db2, float* __restrict__ out) {
    __shared__ float sh[HD];
    __shared__ float st[HD];
    const int n = blockIdx.x;
    const int t = threadIdx.x;
    sh[t] = h[(size_t)n * HD + t];
    __syncthreads();
    float x = db1[t];
    for (int k = 0; k < HD; ++k) x = fmaf(sh[k], dw1[k * HD + t], x);
    st[t] = silu(x);
    __syncthreads();
    if (t < 3) {
        float y = db2[t];
        for (int k = 0; k < HD; ++k) y = fmaf(st[k], dw2[k * 3 + t], y);
        out[(size_t)n * 3 + t] = y;
    }
}

// ---------------------------------------------------------------- launch
extern "C" void kernel_launch(void* const* d_in, const int* in_sizes, int n_in,
                              void* d_out, int out_size, void* d_ws, size_t ws_size,
                              hipStream_t stream) {
    const float* loc       = (const float*)d_in[0];
    const float* vel       = (const float*)d_in[1];
    const float* edge_attr = (const float*)d_in[2];
    const int*   row       = (const int*)d_in[3];
    const int*   col       = (const int*)d_in[4];
    const float* emb_w     = (const float*)d_in[5];
    const float* emb_b     = (const float*)d_in[6];
    const float* ew1       = (const float*)d_in[7];
    const float* eb1       = (const float*)d_in[8];
    const float* ew2       = (const float*)d_in[9];
    const float* eb2       = (const float*)d_in[10];
    const float* nw1       = (const float*)d_in[11];
    const float* nb1       = (const float*)d_in[12];
    const float* nw2       = (const float*)d_in[13];
    const float* nb2       = (const float*)d_in[14];
    const float* dw1       = (const float*)d_in[15];
    const float* db1       = (const float*)d_in[16];
    const float* dw2       = (const float*)d_in[17];
    const float* db2       = (const float*)d_in[18];
    float* out = (float*)d_out;

    float* h   = (float*)d_ws;                 // N*H floats
    float* agg = h + (size_t)NNODES * HD;      // N*H floats

    embed_kernel<<<(NNODES * HD + 255) / 256, 256, 0, stream>>>(loc, vel, emb_w,
                                                                emb_b, h);

    const int edge_tiles = NEDGES / 16;              // 37500 (divisible by 4)
    const int node_tiles = NNODES / 16;              // 3125
    const int edge_blocks = edge_tiles / 4;          // 9375
    const int node_blocks = (node_tiles + 3) / 4;    // 782 (last block clamped)

    for (int i = 0; i < NLAYERS; ++i) {
        zero_kernel<<<(NNODES * HD + 255) / 256, 256, 0, stream>>>(agg,
                                                                   NNODES * HD);
        edge_kernel<<<edge_blocks, 128, 0, stream>>>(
            h, edge_attr, row, col, ew1 + (size_t)i * 258 * HD, eb1 + i * HD,
            ew2 + (size_t)i * HD * HD, eb2 + i * HD, agg);
        node_kernel<<<node_blocks, 128, 0, stream>>>(
            h, agg, nw1 + (size_t)i * 2 * HD * HD, nb1 + i * HD,
            nw2 + (size_t)i * HD * HD, nb2 + i * HD);
    }

    decode_kernel<<<NNODES, 128, 0, stream>>>(h, dw1, db1, dw2, db2, out);
}